// ViewMorphing_15814069584226
// MI455X (gfx1250) — compile-verified
//
#include <hip/hip_runtime.h>

// Problem constants (fixed by the reference: BATCH=32, IMG_DIM=512)
#define DDIM   512
#define HWSZ   (DDIM * DDIM)        // 262144 = 2^18
#define NBATCH 32
#define NPIX   (NBATCH * HWSZ)      // 8,388,608 total pixels
#define OUTPIX (NBATCH * 3 * HWSZ)  // image part of d_out
#define NPART  4096                 // partial-sum slots (one per block)
#define TPB    256

typedef float v2f __attribute__((ext_vector_type(2)));
typedef float v8f __attribute__((ext_vector_type(8)));

// Bilinear sample of one 3-channel image at (qx,qy) with reference semantics.
__device__ __forceinline__ void bilin_sample(const float* __restrict__ img,
                                             float qx, float qy, float m,
                                             float& r0, float& r1, float& r2,
                                             float& lsum) {
  const float lo = 0.001f;
  const float hi = (float)DDIM - 1.001f;  // 510.999f
  float ax = fminf(fmaxf(qx, lo), hi);
  float ay = fminf(fmaxf(qy, lo), hi);
  float dx = qx - ax, dy = qy - ay;
  lsum += dx * dx + dy * dy;  // (q_orig - q_clip)^2, both components

  float fx = floorf(ax), fy = floorf(ay);
  float cx = ceilf(ax),  cy = ceilf(ay);
  float wfx = 1.0f - (ax - fx);
  float wcx = 1.0f - (cx - ax);
  float wfy = 1.0f - (ay - fy);
  float wcy = 1.0f - (cy - ay);

  int ixf = (int)fx, iyf = (int)fy, ixc = (int)cx, iyc = (int)cy;
  int i1 = iyf + (ixf << 9);  // (fx,fy): ind = y + D*x, D=512
  int i2 = iyf + (ixc << 9);  // (cx,fy)
  int i3 = iyc + (ixf << 9);  // (fx,cy)
  int i4 = iyc + (ixc << 9);  // (cx,cy)
  float w1 = wfx * wfy, w2 = wcx * wfy, w3 = wfx * wcy, w4 = wcx * wcy;

  const float* __restrict__ p0 = img;
  const float* __restrict__ p1 = img + HWSZ;
  const float* __restrict__ p2 = img + 2 * HWSZ;
  r0 = m * (w1 * p0[i1] + w2 * p0[i2] + w3 * p0[i3] + w4 * p0[i4]);
  r1 = m * (w1 * p1[i1] + w2 * p1[i2] + w3 * p1[i3] + w4 * p1[i4]);
  r2 = m * (w1 * p2[i1] + w2 * p2[i2] + w3 * p2[i3] + w4 * p2[i4]);
}

__global__ __launch_bounds__(TPB)
void vm_warp_kernel(const float* __restrict__ im1, const float* __restrict__ im2,
                    const float* __restrict__ C,   const float* __restrict__ M1,
                    const float* __restrict__ M2,  float* __restrict__ out,
                    float* __restrict__ partials) {
  const int tid    = blockIdx.x * TPB + threadIdx.x;
  const int stride = NPART * TPB;  // 1,048,576 threads -> 8 pixels/thread
  float lsum = 0.0f;

  for (int idx = tid; idx < NPIX; idx += stride) {
    int n = idx >> 18;           // idx / HWSZ
    int p = idx & (HWSZ - 1);    // idx % HWSZ
    int cbase = (n << 19) + p;   // n*2*HWSZ + p

    // Prefetch next grid-stride iteration's streaming inputs (global_prefetch_b8).
    __builtin_prefetch(C  + cbase + stride, 0, 0);
    __builtin_prefetch(M1 + idx   + stride, 0, 0);
    __builtin_prefetch(M2 + idx   + stride, 0, 0);

    float c0 = C[cbase];           // C[n,0,p]
    float c1 = C[cbase + HWSZ];    // C[n,1,p]
    float m1 = M1[idx];
    float m2 = M2[idx];

    float gx = (float)(p >> 9);    // q[0] = row
    float gy = (float)(p & 511);   // q[1] = col

    const float* imb1 = im1 + n * 3 * HWSZ;
    const float* imb2 = im2 + n * 3 * HWSZ;

    float a0, a1, a2, b0, b1, b2;
    bilin_sample(imb1, gx + c0, gy + c1, m1, a0, a1, a2, lsum);
    bilin_sample(imb2, gx - c0, gy - c1, m2, b0, b1, b2, lsum);

    int obase = n * 3 * HWSZ + p;
    out[obase]            = a0 + b0;
    out[obase + HWSZ]     = a1 + b1;
    out[obase + 2 * HWSZ] = a2 + b2;
  }

  // Deterministic block reduction of the loss contribution (wave32).
  for (int o = 16; o > 0; o >>= 1) lsum += __shfl_down(lsum, o, 32);
  __shared__ float sm[TPB / 32];
  int wid = threadIdx.x >> 5, ln = threadIdx.x & 31;
  if (ln == 0) sm[wid] = lsum;
  __syncthreads();
  if (threadIdx.x < (TPB / 32)) {
    float v = sm[threadIdx.x];
    for (int o = (TPB / 64); o > 0; o >>= 1) v += __shfl_down(v, o, 32);
    if (threadIdx.x == 0) partials[blockIdx.x] = v;
  }
}

// Reduce NPART partials with V_WMMA_F32_16X16X4_F32.
// A = all-ones (16x4)  =>  D[i][j] = sum_k B[k][j] + C[i][j].
// Every element loaded into B is counted exactly once in sum_j D[0][j],
// regardless of B's lane layout, so the reduction is layout-robust.
__global__ __launch_bounds__(32)
void vm_loss_reduce_wmma(const float* __restrict__ partials,
                         float* __restrict__ loss_out, float scale) {
  int lane = threadIdx.x;  // one wave32, EXEC all ones (WMMA requirement)
  v8f acc = {};
  v2f a;
  a[0] = 1.0f; a[1] = 1.0f;
  for (int base = 0; base < NPART; base += 64) {
    v2f b;
    b[0] = partials[base + 2 * lane];
    b[1] = partials[base + 2 * lane + 1];
    acc = __builtin_amdgcn_wmma_f32_16x16x4_f32(
        /*neg_a=*/false, a, /*neg_b=*/false, b,
        /*c_mod=*/(short)0, acc, /*reuse_a=*/false, /*reuse_b=*/false);
  }
  // C/D layout: VGPR0 lanes 0..15 = D[0][0..15]; sum row 0 across 16 lanes.
  float v = acc[0];
  v += __shfl_xor(v, 1, 32);
  v += __shfl_xor(v, 2, 32);
  v += __shfl_xor(v, 4, 32);
  v += __shfl_xor(v, 8, 32);
  if (lane == 0) loss_out[0] = v * scale;
}

extern "C" void kernel_launch(void* const* d_in, const int* in_sizes, int n_in,
                              void* d_out, int out_size, void* d_ws, size_t ws_size,
                              hipStream_t stream) {
  const float* im1 = (const float*)d_in[0];
  const float* im2 = (const float*)d_in[1];
  const float* C   = (const float*)d_in[2];
  const float* M1  = (const float*)d_in[3];
  const float* M2  = (const float*)d_in[4];
  float* out       = (float*)d_out;
  float* partials  = (float*)d_ws;  // NPART floats = 16 KB

  vm_warp_kernel<<<NPART, TPB, 0, stream>>>(im1, im2, C, M1, M2, out, partials);

  // loss = (sum_a + sum_b) * 0.01 / (N*2*HW) / (D*D); N*2*HW*HW = 2^42 (exact)
  float scale = (float)(0.01 / ((double)NBATCH * 2.0 * (double)HWSZ * (double)HWSZ));
  vm_loss_reduce_wmma<<<1, 32, 0, stream>>>(partials, out + OUTPIX, scale);
}